// GradientMagnitude_18897856102902
// MI455X (gfx1250) — compile-verified
//
#include <hip/hip_runtime.h>
#include <hip/hip_bf16.h>
#include <math.h>

typedef float v2f __attribute__((ext_vector_type(2)));
typedef float v8f __attribute__((ext_vector_type(8)));

#define IMG_H 512
#define IMG_W 512
#define NBATCH 64
#define TY 16
#define TX 128
#define IN_STRIDE 161   // >= 160 needed; odd -> conflict-light A loads
#define HB_STRIDE 180   // 2*180%64=40, 8*180%64=32 -> conflict-free B loads / C stores
#define NTHREADS 256

// 1-D normalized Gaussian, sigma=5, ks=15 (2-D kernel = outer product of this)
__constant__ float cW15[15] = {
    0.03453788f, 0.04479321f, 0.05581578f, 0.06682362f, 0.07686546f,
    0.08494946f, 0.09020245f, 0.09202463f, 0.09020245f, 0.08494946f,
    0.07686546f, 0.06682362f, 0.05581578f, 0.04479321f, 0.03453788f};

__global__ __launch_bounds__(NTHREADS)
void gm_tile_kernel(const float* __restrict__ x, float* __restrict__ gmout,
                    unsigned int* __restrict__ minb, unsigned int* __restrict__ maxb)
{
    __shared__ float sIn[32 * IN_STRIDE];    // input tile (rows y0-8..y0+23, cols x0-8..x0+152)
    __shared__ float sHb[32 * HB_STRIDE];    // horizontally blurred (cols n=7..150 used)
    __shared__ float sVb[18 * HB_STRIDE];    // fully blurred (rows y0-1..y0+16)
    __shared__ float sBand[32 * 16];         // Band[j][n] = w[j-n] (band 0..14)
    __shared__ float sRmin[NTHREADS];
    __shared__ float sRmax[NTHREADS];

    const int tid = threadIdx.x;
    const int bz = blockIdx.z;
    const int y0 = blockIdx.y * TY;
    const int x0 = blockIdx.x * TX;
    const float* img = x + (size_t)bz * IMG_H * IMG_W;

    // --- build band matrix in LDS ---
    for (int i = tid; i < 512; i += NTHREADS) {
        int j = i >> 4, n = i & 15, d = j - n;
        sBand[i] = (d >= 0 && d <= 14) ? cW15[d] : 0.0f;
    }
    // --- load zero-padded input tile ---
    for (int i = tid; i < 32 * IN_STRIDE; i += NTHREADS) {
        int r = i / IN_STRIDE, c = i - r * IN_STRIDE;
        int gy = y0 - 8 + r, gx = x0 - 8 + c;
        float v = 0.0f;
        if ((unsigned)gy < IMG_H && (unsigned)gx < IMG_W) v = img[gy * IMG_W + gx];
        sIn[i] = v;
    }
    __syncthreads();

    const int wave  = tid >> 5;
    const int lane  = tid & 31;
    const int mrow  = lane & 15;   // M (A) or N (B/C) index for this lane
    const int khalf = lane >> 4;   // 0: K{0,1} / M{0..7}; 1: K{2,3} / M{8..15}

    // ---------------- horizontal blur:  Out(16x16) = In(16x32) x Band(32x16) ----------------
    for (int t = wave; t < 18; t += 8) {
        int rg = t / 9, ct = t - rg * 9;
        int row0 = rg * 16;
        int cin  = ct * 16;        // input col base of 32-wide K window
        int cout = 7 + ct * 16;    // hb col base
        v8f acc = {};
        #pragma unroll
        for (int s = 0; s < 8; ++s) {
            int jr = s * 4 + khalf * 2;
            v2f a, b;
            a.x = sIn[(row0 + mrow) * IN_STRIDE + cin + jr];
            a.y = sIn[(row0 + mrow) * IN_STRIDE + cin + jr + 1];
            b.x = sBand[jr * 16 + mrow];
            b.y = sBand[(jr + 1) * 16 + mrow];
            acc = __builtin_amdgcn_wmma_f32_16x16x4_f32(false, a, false, b,
                                                        (short)0, acc, false, false);
        }
        #pragma unroll
        for (int v = 0; v < 8; ++v)
            sHb[(row0 + v + 8 * khalf) * HB_STRIDE + cout + mrow] = acc[v];
    }
    __syncthreads();

    // ---------------- vertical blur:  Out(16x16) = Band^T(16x32) x Hb(32x16) ----------------
    for (int t = wave; t < 18; t += 8) {
        int rg = t / 9, ct = t - rg * 9;
        int cb = 7 + ct * 16;
        v8f acc = {};
        #pragma unroll
        for (int s = 0; s < 8; ++s) {
            int jr = s * 4 + khalf * 2;
            v2f a, b;
            if (rg == 0) {                 // out rows r'=7+m  -> A[m][j]=w[j-m]
                a.x = sBand[jr * 16 + mrow];
                a.y = sBand[(jr + 1) * 16 + mrow];
            } else {                       // out rows r'=9+m  -> A[m][j]=w[j-2-m]
                a.x = (jr >= 2) ? sBand[(jr - 2) * 16 + mrow] : 0.0f;
                a.y = (jr >= 1) ? sBand[(jr - 1) * 16 + mrow] : 0.0f;
            }
            b.x = sHb[jr * HB_STRIDE + cb + mrow];
            b.y = sHb[(jr + 1) * HB_STRIDE + cb + mrow];
            acc = __builtin_amdgcn_wmma_f32_16x16x4_f32(false, a, false, b,
                                                        (short)0, acc, false, false);
        }
        #pragma unroll
        for (int v = 0; v < 8; ++v) {
            int m = v + 8 * khalf;
            if (rg == 0) {
                sVb[m * HB_STRIDE + cb + mrow] = acc[v];            // rows y0-1 .. y0+14
            } else if (m >= 14) {
                sVb[(m + 2) * HB_STRIDE + cb + mrow] = acc[v];      // rows y0+15, y0+16
            }
        }
    }
    __syncthreads();

    // ---------------- Sobel + magnitude + per-block min/max ----------------
    float lmin = 3.402823466e38f, lmax = 0.0f;
    for (int i = tid; i < TY * TX; i += NTHREADS) {
        int r = i >> 7, c = i & 127;
        int gy = y0 + r, gx = x0 + c;
        float p[3][3];
        #pragma unroll
        for (int dy = -1; dy <= 1; ++dy)
            #pragma unroll
            for (int dx = -1; dx <= 1; ++dx) {
                int yy = gy + dy, xx = gx + dx;
                p[dy + 1][dx + 1] = ((unsigned)yy < IMG_H && (unsigned)xx < IMG_W)
                    ? sVb[(r + 1 + dy) * HB_STRIDE + 8 + c + dx] : 0.0f;
            }
        float sx = (p[0][2] + 2.0f * p[1][2] + p[2][2]) - (p[0][0] + 2.0f * p[1][0] + p[2][0]);
        float sy = (p[2][0] + 2.0f * p[2][1] + p[2][2]) - (p[0][0] + 2.0f * p[0][1] + p[0][2]);
        float g = sqrtf(sx * sx + sy * sy + 1e-6f);
        gmout[((size_t)bz * IMG_H + gy) * IMG_W + gx] = g;
        lmin = fminf(lmin, g);
        lmax = fmaxf(lmax, g);
    }
    sRmin[tid] = lmin; sRmax[tid] = lmax;
    __syncthreads();
    for (int s2 = NTHREADS / 2; s2 > 0; s2 >>= 1) {
        if (tid < s2) {
            sRmin[tid] = fminf(sRmin[tid], sRmin[tid + s2]);
            sRmax[tid] = fmaxf(sRmax[tid], sRmax[tid + s2]);
        }
        __syncthreads();
    }
    if (tid == 0) {
        // gm >= 0 -> IEEE bits are monotone as unsigned
        atomicMin(&minb[bz], __float_as_uint(sRmin[0]));
        atomicMax(&maxb[bz], __float_as_uint(sRmax[0]));
    }
}

__global__ void init_minmax(unsigned int* __restrict__ minb, unsigned int* __restrict__ maxb)
{
    int t = threadIdx.x;
    if (t < NBATCH) { minb[t] = 0x7f800000u; maxb[t] = 0u; }
}

__global__ __launch_bounds__(NTHREADS)
void normalize_kernel(float* __restrict__ out,
                      const unsigned int* __restrict__ minb,
                      const unsigned int* __restrict__ maxb)
{
    size_t i = ((size_t)blockIdx.x * NTHREADS + threadIdx.x) * 4;
    int b = (int)(i >> 18);   // 512*512 = 2^18 elements per batch
    float mn = __uint_as_float(minb[b]);
    float mx = __uint_as_float(maxb[b]);
    float inv = 1.0f / (mx - mn + 1e-6f);
    float4 v = *reinterpret_cast<float4*>(out + i);
    v.x = (v.x - mn) * inv;
    v.y = (v.y - mn) * inv;
    v.z = (v.z - mn) * inv;
    v.w = (v.w - mn) * inv;
    *reinterpret_cast<float4*>(out + i) = v;
}

extern "C" void kernel_launch(void* const* d_in, const int* in_sizes, int n_in,
                              void* d_out, int out_size, void* d_ws, size_t ws_size,
                              hipStream_t stream)
{
    (void)in_sizes; (void)n_in; (void)out_size; (void)ws_size;
    const float* x = (const float*)d_in[0];
    float* out = (float*)d_out;
    unsigned int* minb = (unsigned int*)d_ws;
    unsigned int* maxb = minb + NBATCH;

    init_minmax<<<1, NBATCH, 0, stream>>>(minb, maxb);

    dim3 grid(IMG_W / TX, IMG_H / TY, NBATCH);   // (4, 32, 64)
    gm_tile_kernel<<<grid, NTHREADS, 0, stream>>>(x, out, minb, maxb);

    const size_t total = (size_t)NBATCH * IMG_H * IMG_W;
    normalize_kernel<<<(unsigned)(total / (NTHREADS * 4)), NTHREADS, 0, stream>>>(out, minb, maxb);
}